// WaveletMixing2D_38955353374743
// MI455X (gfx1250) — compile-verified
//
#include <hip/hip_runtime.h>

#define TPB 256
#define LD64 65
#define LD32 33
#define IMGS_PER_BLOCK 2

typedef float v2f __attribute__((ext_vector_type(2)));
typedef float v8f __attribute__((ext_vector_type(8)));

// db4 orthonormal scaling filter (pywt rec_lo). g[i] = (-1)^i * h[7-i].
__constant__ float c_H[8] = {
    0.23037781330885523f,  0.7148465705525415f,   0.6308807679295904f,
    -0.02798376941698385f, -0.18703481171888114f, 0.030841381835986965f,
    0.032883011666982945f, -0.010597401784997278f};

// ---- Structural sparsity of the periodic analysis matrix ----------------
// A_N[k][j] != 0  <=>  (j - 2*(k mod N/2)) mod N in [0,8)   (8-tap db4 band)
constexpr bool anz(int N, int k, int j) {
  int kk = (k >= N / 2) ? (k - N / 2) : k;
  int d = (j - 2 * kk) % N;
  if (d < 0) d += N;
  return d < 8;
}
// K-tile jt needed when A rows r0..r0+15 index K directly (A[r][k] pattern)
constexpr unsigned mask_rows(int N, int r0) {
  unsigned m = 0;
  for (int jt = 0; jt < N / 4; ++jt)
    for (int k = 4 * jt; k < 4 * jt + 4; ++k)
      for (int r = r0; r < r0 + 16; ++r)
        if (anz(N, r, k)) m |= 1u << jt;
  return m;
}
// K-tile jt needed when K indexes A rows (A[k][c] pattern, i.e. A^T operand)
constexpr unsigned mask_cols(int N, int c0) {
  unsigned m = 0;
  for (int jt = 0; jt < N / 4; ++jt)
    for (int k = 4 * jt; k < 4 * jt + 4; ++k)
      for (int c = c0; c < c0 + 16; ++c)
        if (anz(N, k, c)) m |= 1u << jt;
  return m;
}
// Mask selector (all constexpr -> folds at template instantiation):
// SEL 0: A left (normal)      -> rows-mask by row block
// SEL 1: A^T right            -> rows-mask by col block
// SEL 2: A^T left             -> cols-mask by row block
// SEL 3: A right (normal)     -> cols-mask by col block
constexpr unsigned sel_mask(int N, int SEL, int rb, int cb) {
  int blk = (SEL == 0 || SEL == 2) ? rb : cb;
  return (SEL <= 1) ? mask_rows(N, 16 * blk) : mask_cols(N, 16 * blk);
}

// Local integer sequence (avoid <utility> in device code)
template <int... Is> struct iseq {};
template <int N, int... Is> struct mkseq : mkseq<N - 1, N - 1, Is...> {};
template <int... Is> struct mkseq<0, Is...> { using type = iseq<Is...>; };

// One 16x16 f32 tile of P = op(M1) * op(M2) via V_WMMA_F32_16X16X4_F32.
// `mask` is a compile-time constant at every call site: guards fold away,
// leaving straight-line batched ds_load + wmma clauses.
template <int K, bool TA, bool TB>
__device__ inline v8f wmma_tile(const float* __restrict__ M1, int ld1,
                                const float* __restrict__ M2, int ld2,
                                int r0, int c0, int lane, unsigned mask) {
  v8f acc = {0.f, 0.f, 0.f, 0.f, 0.f, 0.f, 0.f, 0.f};
  const int hw = lane >> 4;       // half-wave: K pairs 0-1 vs 2-3
  const int l16 = lane & 15;
  const int row = r0 + l16;       // A rows: both half-waves hold M=0..15
  const int col = c0 + l16;       // B cols striped across lanes
#pragma unroll
  for (int j = 0; j < K / 4; ++j) {
    if (mask & (1u << j)) {       // constant-folded after unrolling
      const int ka = 4 * j + 2 * hw;
      v2f a, b;
      if (TA) { a.x = M1[ka * ld1 + row];  a.y = M1[(ka + 1) * ld1 + row]; }
      else    { a.x = M1[row * ld1 + ka];  a.y = M1[row * ld1 + ka + 1];   }
      if (TB) { b.x = M2[col * ld2 + ka];  b.y = M2[col * ld2 + ka + 1];   }
      else    { b.x = M2[ka * ld2 + col];  b.y = M2[(ka + 1) * ld2 + col]; }
      acc = __builtin_amdgcn_wmma_f32_16x16x4_f32(false, a, false, b,
                                                  (short)0, acc, false, false);
    }
  }
  return acc;
}

// Scatter C/D tile per ISA layout: VGPR v holds row M = v + 8*(lane>=16).
__device__ inline void store_tile(float* __restrict__ D, int ld, int r0, int c0,
                                  int lane, v8f acc) {
  const int hw = lane >> 4;
  const int c = c0 + (lane & 15);
#pragma unroll
  for (int v = 0; v < 8; ++v) D[(r0 + v + 8 * hw) * ld + c] = acc[v];
}

// ---- 64x64 product stage: 16 tiles, tiles T and T+8 -> wave (T&7) -------
template <bool TA, bool TB, int SEL>
struct S64 {
  template <int T>
  static __device__ inline void one(const float* M1, const float* M2, float* D,
                                    int wave, int lane) {
    constexpr int r0 = (T >> 2) << 4, c0 = (T & 3) << 4;
    constexpr unsigned mask = sel_mask(64, SEL, T >> 2, T & 3);
    if (wave == (T & 7)) {        // scalar, uniform: EXEC stays all-ones
      v8f acc = wmma_tile<64, TA, TB>(M1, LD64, M2, LD64, r0, c0, lane, mask);
      store_tile(D, LD64, r0, c0, lane, acc);
    }
  }
  template <int... Ts>
  static __device__ inline void go(const float* M1, const float* M2, float* D,
                                   int wave, int lane, iseq<Ts...>) {
    (one<Ts>(M1, M2, D, wave, lane), ...);
  }
  static __device__ inline void run(const float* M1, const float* M2, float* D,
                                    int wave, int lane) {
    go(M1, M2, D, wave, lane, typename mkseq<16>::type{});
  }
};

// ---- 32x32 product stage: 4 tiles -> waves 0..3 -------------------------
template <bool TA, bool TB, int SEL>
struct S32 {
  template <int T>
  static __device__ inline void one(const float* M1, int ld1, const float* M2,
                                    int ld2, float* D, int ldd, int wave, int lane) {
    constexpr int r0 = (T >> 1) << 4, c0 = (T & 1) << 4;
    constexpr unsigned mask = sel_mask(32, SEL, T >> 1, T & 1);
    if (wave == T) {
      v8f acc = wmma_tile<32, TA, TB>(M1, ld1, M2, ld2, r0, c0, lane, mask);
      store_tile(D, ldd, r0, c0, lane, acc);
    }
  }
  template <int... Ts>
  static __device__ inline void go(const float* M1, int ld1, const float* M2,
                                   int ld2, float* D, int ldd, int wave, int lane,
                                   iseq<Ts...>) {
    (one<Ts>(M1, ld1, M2, ld2, D, ldd, wave, lane), ...);
  }
  static __device__ inline void run(const float* M1, int ld1, const float* M2,
                                    int ld2, float* D, int ldd, int wave, int lane) {
    go(M1, ld1, M2, ld2, D, ldd, wave, lane, typename mkseq<4>::type{});
  }
};

// ---- Final stage: recon tile * A64, fuse residual add + global store ----
template <int T>
__device__ inline void final_tile(const float* sCt, const float* sA,
                                  const float* sXi, float* om, int wave, int lane) {
  constexpr int r0 = (T >> 2) << 4, c0 = (T & 3) << 4;
  constexpr unsigned mask = sel_mask(64, 3, T >> 2, T & 3);
  if (wave == (T & 7)) {
    v8f acc = wmma_tile<64, false, false>(sCt, LD64, sA, LD64, r0, c0, lane, mask);
    const int hw = lane >> 4;
    const int c = c0 + (lane & 15);
#pragma unroll
    for (int v = 0; v < 8; ++v) {
      const int r = r0 + v + 8 * hw;
      om[r * 64 + c] = sXi[r * LD64 + c] + acc[v];
    }
  }
}
template <int... Ts>
__device__ inline void final_go(const float* sCt, const float* sA, const float* sXi,
                                float* om, int wave, int lane, iseq<Ts...>) {
  (final_tile<Ts>(sCt, sA, sXi, om, wave, lane), ...);
}

// Zero-padded 3x3 conv on an n x n LDS band: dst = relu?((conv+bias)*scale+shift)
__device__ inline void conv_band(const float* __restrict__ src, int sld, int sr, int sc,
                                 float* __restrict__ dst, int dld, int dr, int dc,
                                 int n, const float* __restrict__ wg, float bias,
                                 float scale, float shift, bool relu, int tid) {
  float w[9];
#pragma unroll
  for (int q = 0; q < 9; ++q) w[q] = wg[q];
  for (int p = tid; p < n * n; p += TPB) {
    const int i = p / n, j = p - i * n;
    float s = 0.f;
#pragma unroll
    for (int di = 0; di < 3; ++di) {
      const int ii = i + di - 1;
      if (ii < 0 || ii >= n) continue;
#pragma unroll
      for (int dj = 0; dj < 3; ++dj) {
        const int jj = j + dj - 1;
        if (jj < 0 || jj >= n) continue;
        s += w[di * 3 + dj] * src[(sr + ii) * sld + (sc + jj)];
      }
    }
    float y = (s + bias) * scale + shift;
    if (relu) y = fmaxf(y, 0.f);
    dst[(dr + i) * dld + (dc + j)] = y;
  }
}

__global__ __launch_bounds__(TPB)
void wavelet_mix2d_kernel(const float* __restrict__ x,
                          const float* __restrict__ ll_w,
                          const float* __restrict__ ll_b,
                          const float* __restrict__ bn_gamma,
                          const float* __restrict__ bn_beta,
                          const float* __restrict__ det_w,
                          const float* __restrict__ det_b,
                          float* __restrict__ out, int n_images) {
  __shared__ float sX[64 * LD64];    // input image (kept for residual)
  __shared__ float sT[64 * LD64];    // temp / assembled T1'
  __shared__ float sC[64 * LD64];    // T1 coeffs, later inverse temp
  __shared__ float sA64[64 * LD64];  // 64x64 analysis matrix
  __shared__ float sA32[32 * LD32];  // 32x32 analysis matrix
  __shared__ float s2a[32 * LD32];
  __shared__ float s2c[32 * LD32];   // T2 coeffs
  __shared__ float s2m[32 * LD32];   // mixed T2'

  const int tid = threadIdx.x;
  const int lane = tid & 31;
  // Force wave id scalar so all tile guards are uniform scalar branches.
  const int wave = __builtin_amdgcn_readfirstlane(tid >> 5);

  // ---- Build analysis matrices once per block ---------------------------
  for (int i = tid; i < 64 * LD64; i += TPB) sA64[i] = 0.f;
  for (int i = tid; i < 32 * LD32; i += TPB) sA32[i] = 0.f;
  __syncthreads();
  for (int idx = tid; idx < 512; idx += TPB) {           // 64 rows * 8 taps
    const int k = idx >> 3, i = idx & 7;
    const bool hi = k >= 32;
    const float f = hi ? (((i & 1) ? -1.f : 1.f) * c_H[7 - i]) : c_H[i];
    const int kk = hi ? (k - 32) : k;
    sA64[k * LD64 + ((2 * kk + i) & 63)] = f;
  }
  {                                                      // 32 rows * 8 taps
    const int k = tid >> 3, i = tid & 7;                 // tid covers 0..255
    const bool hi = k >= 16;
    const float f = hi ? (((i & 1) ? -1.f : 1.f) * c_H[7 - i]) : c_H[i];
    const int kk = hi ? (k - 16) : k;
    sA32[k * LD32 + ((2 * kk + i) & 31)] = f;
  }
  const float bn_scale = bn_gamma[0] * rsqrtf(1.f + 1e-5f);

  for (int it = 0; it < IMGS_PER_BLOCK; ++it) {
    const int m = blockIdx.x * IMGS_PER_BLOCK + it;
    if (m >= n_images) break;                            // block-uniform
    const size_t img = (size_t)m * 4096;
    const float* xm = x + img;
    __syncthreads();                                     // sX/sC reusable
    for (int i = tid; i < 4096; i += TPB) sX[(i >> 6) * LD64 + (i & 63)] = xm[i];
    __syncthreads();

    // S1: sT = A64 * X
    S64<false, false, 0>::run(sA64, sX, sT, wave, lane);
    __syncthreads();
    // S2: sC = sT * A64^T   (T1 quadrants: ll|hl / lh|hh)
    S64<false, true, 1>::run(sT, sA64, sC, wave, lane);
    __syncthreads();
    // S3a: s2a = A32 * ll1  (ll1 = sC[0:32,0:32])
    S32<false, false, 0>::run(sA32, LD32, sC, LD64, s2a, LD32, wave, lane);
    __syncthreads();
    // S3b: s2c = s2a * A32^T
    S32<false, true, 1>::run(s2a, LD32, sA32, LD32, s2c, LD32, wave, lane);
    __syncthreads();
    // S4: mix level-2 bands into s2m
    conv_band(s2c, LD32, 0, 0,   s2m, LD32, 0, 0,   16, ll_w,          ll_b[0],
              bn_scale, bn_beta[0], true, tid);                       // ll2
    conv_band(s2c, LD32, 16, 0,  s2m, LD32, 16, 0,  16, det_w + 3 * 9, det_b[3],
              1.f, 0.f, false, tid);                                  // lh2
    conv_band(s2c, LD32, 0, 16,  s2m, LD32, 0, 16,  16, det_w + 4 * 9, det_b[4],
              1.f, 0.f, false, tid);                                  // hl2
    conv_band(s2c, LD32, 16, 16, s2m, LD32, 16, 16, 16, det_w + 5 * 9, det_b[5],
              1.f, 0.f, false, tid);                                  // hh2
    __syncthreads();
    // S5a: s2a = A32^T * s2m
    S32<true, false, 2>::run(sA32, LD32, s2m, LD32, s2a, LD32, wave, lane);
    __syncthreads();
    // S5b: sT[0:32,0:32] = s2a * A32   (ll1')
    S32<false, false, 3>::run(s2a, LD32, sA32, LD32, sT, LD64, wave, lane);
    __syncthreads();
    // S6: mix level-1 detail bands: sC -> sT
    conv_band(sC, LD64, 32, 0,  sT, LD64, 32, 0,  32, det_w + 0 * 9, det_b[0],
              1.f, 0.f, false, tid);                                  // lh1
    conv_band(sC, LD64, 0, 32,  sT, LD64, 0, 32,  32, det_w + 1 * 9, det_b[1],
              1.f, 0.f, false, tid);                                  // hl1
    conv_band(sC, LD64, 32, 32, sT, LD64, 32, 32, 32, det_w + 2 * 9, det_b[2],
              1.f, 0.f, false, tid);                                  // hh1
    __syncthreads();
    // S7a: sC = A64^T * sT
    S64<true, false, 2>::run(sA64, sT, sC, wave, lane);
    __syncthreads();
    // S7b: out = x + sC * A64
    final_go(sC, sA64, sX, out + img, wave, lane, typename mkseq<16>::type{});
  }
}

extern "C" void kernel_launch(void* const* d_in, const int* in_sizes, int n_in,
                              void* d_out, int out_size, void* d_ws, size_t ws_size,
                              hipStream_t stream) {
  const float* x        = (const float*)d_in[0];
  const float* ll_w     = (const float*)d_in[1];
  const float* ll_b     = (const float*)d_in[2];
  const float* bn_gamma = (const float*)d_in[3];
  const float* bn_beta  = (const float*)d_in[4];
  const float* det_w    = (const float*)d_in[5];
  const float* det_b    = (const float*)d_in[6];
  float* out = (float*)d_out;
  const int n_images = in_sizes[0] / 4096;  // B*C = 8192 images of 64x64
  const int blocks = (n_images + IMGS_PER_BLOCK - 1) / IMGS_PER_BLOCK;
  wavelet_mix2d_kernel<<<dim3(blocks), dim3(TPB), 0, stream>>>(
      x, ll_w, ll_b, bn_gamma, bn_beta, det_w, det_b, out, n_images);
}